// DGCNN_38860864094769
// MI455X (gfx1250) — compile-verified
//
#include <hip/hip_runtime.h>
#include <math.h>

#define N_GRAPHS 500
#define N_PER    200
#define DEG_E    16
#define EPG      (N_PER * DEG_E)   // 3200 edges per graph (contiguous block)
#define N_NODES  (N_GRAPHS * N_PER)
#define FIN      128
#define HDIM     32
#define K_TOP    30
#define CAT      97
#define NT       256               // 8 waves of 32
#define MPAD     208               // 13 * 16 (M padded for 16x16 WMMA tiles)
#define HS       36                // feature row stride (16B aligned, bank-spread)

typedef float v2f __attribute__((ext_vector_type(2)));
typedef float v8f __attribute__((ext_vector_type(8)));

// C = A(row-major, lda) @ B(K x 32, row-major) -> out (LDS, row stride HS).
// V_WMMA_F32_16X16X4_F32; 26 tiles (13 M x 2 N) across 8 waves; tile index is
// wave-uniform so EXEC stays all-ones through every WMMA (ISA requirement).
__device__ __forceinline__ void gemm_tiles(const float* __restrict__ A, int lda,
                                           long long rowBase, long long rowMax,
                                           int K, const float* __restrict__ B,
                                           float* __restrict__ out, int tid) {
  const int wave = tid >> 5;
  const int lane = tid & 31;
  const int l15  = lane & 15;
  const int koff = (lane < 16) ? 0 : 2;   // A/B K-split across lane halves
  for (int tile = wave; tile < 26; tile += 8) {
    const int Mbase = (tile >> 1) * 16;
    const int Nbase = (tile & 1) * 16;
    long long grow = rowBase + Mbase + l15;
    if (grow > rowMax) grow = rowMax;     // pad rows: harmless, D rows discarded
    const float* arow = A + grow * (long long)lda;
    const int col = Nbase + l15;
    v8f c = {0.f, 0.f, 0.f, 0.f, 0.f, 0.f, 0.f, 0.f};
    for (int k0 = 0; k0 < K; k0 += 4) {
      v2f a, b;
      a.x = arow[k0 + koff];
      a.y = arow[k0 + koff + 1];
      b.x = B[(k0 + koff) * HDIM + col];
      b.y = B[(k0 + koff + 1) * HDIM + col];
      c = __builtin_amdgcn_wmma_f32_16x16x4_f32(false, a, false, b,
                                                (short)0, c, false, false);
    }
    const int rbase = Mbase + ((lane < 16) ? 0 : 8);
#pragma unroll
    for (int r = 0; r < 8; ++r)
      out[(rbase + r) * HS + col] = c[r];
  }
}

// Gather-form GCN propagation (CSR, no atomics):
//   h[i,:] = tanh( invs[i] * sum_{s in adj(i)} t[s,:]*invs[s]
//                  + t[i,:]*invs[i]^2 + bias )
__device__ __forceinline__ void propagate32(const float* __restrict__ t,
                                            float* __restrict__ h,
                                            const float* __restrict__ invs,
                                            const int* __restrict__ offs,
                                            const int* __restrict__ adj,
                                            const float* __restrict__ bias,
                                            int tid) {
  for (int task = tid; task < N_PER * 8; task += NT) {
    const int i  = task >> 3;
    const int cg = (task & 7) * 4;        // 4-channel group, 16B aligned
    float a0 = 0.f, a1 = 0.f, a2 = 0.f, a3 = 0.f;
    const int beg = offs[i], end = offs[i + 1];
    for (int k = beg; k < end; ++k) {
      const int s = adj[k];
      const float w = invs[s];
      const float4 v = *reinterpret_cast<const float4*>(t + s * HS + cg);
      a0 += v.x * w; a1 += v.y * w; a2 += v.z * w; a3 += v.w * w;
    }
    const float wi = invs[i];
    const float ws = wi * wi;             // self-loop weight
    const float* ti = t + i * HS + cg;
    float* hi = h + i * HS + cg;
    hi[0] = tanhf(a0 * wi + ti[0] * ws + bias[cg + 0]);
    hi[1] = tanhf(a1 * wi + ti[1] * ws + bias[cg + 1]);
    hi[2] = tanhf(a2 * wi + ti[2] * ws + bias[cg + 2]);
    hi[3] = tanhf(a3 * wi + ti[3] * ws + bias[cg + 3]);
  }
  __syncthreads();
}

__global__ __launch_bounds__(NT)
void dgcnn_fused(const float* __restrict__ x,
                 const int* __restrict__ esrc, const int* __restrict__ edst,
                 const float* __restrict__ W1, const float* __restrict__ b1,
                 const float* __restrict__ W2, const float* __restrict__ b2,
                 const float* __restrict__ W3, const float* __restrict__ b3,
                 const float* __restrict__ W4, const float* __restrict__ b4,
                 const float* __restrict__ cw1, const float* __restrict__ cb1,
                 const float* __restrict__ cw2, const float* __restrict__ cb2,
                 const float* __restrict__ lw1, const float* __restrict__ lb1,
                 const float* __restrict__ lw2, const float* __restrict__ lb2,
                 float* __restrict__ out) {
  const int g = blockIdx.x;
  const int tid = threadIdx.x;

  extern __shared__ float smem[];
  float* t    = smem;                      // 208*36 transform buffer
  float* h1   = t  + MPAD * HS;            // 208*36
  float* h2   = h1 + MPAD * HS;            // 208*36
  float* h3   = h2 + MPAD * HS;            // 208*36
  float* h4   = h3 + MPAD * HS;            // 208
  float* t4   = h4 + MPAD;                 // 208
  float* invs = t4 + MPAD;                 // 208
  int*   offs = (int*)(invs + MPAD);       // 208 (201 used): CSR row offsets
  int*   cur  = offs + MPAD;               // 208: counts, then fill cursors
  int*   adj  = cur + MPAD;                // 3200: sources bucketed by dst
  int*   topidx = adj + EPG;               // 32
  float* seq  = (float*)(topidx + 32);     // 30*97
  float* c1   = seq + K_TOP * CAT;         // 16*30
  float* pl   = c1 + 16 * 30;              // 16*15
  float* fl   = pl + 16 * 15;              // 352
  float* zz   = fl + 352;                  // 128

  const long long ebase = (long long)g * EPG;
  const int nbase = g * N_PER;

  // ---- in-degree counts; zero WMMA pad rows of h buffers ----
  for (int i = tid; i < N_PER; i += NT) cur[i] = 0;
  for (int i = tid; i < 8 * HS; i += NT) {          // rows 200..207
    h1[N_PER * HS + i] = 0.f;
    h2[N_PER * HS + i] = 0.f;
    h3[N_PER * HS + i] = 0.f;
  }
  __syncthreads();
  for (int e = tid; e < EPG; e += NT)
    atomicAdd(&cur[edst[ebase + e] - nbase], 1);
  __syncthreads();
  // ---- serial prefix sum (200 elements, negligible) ----
  if (tid == 0) {
    int acc = 0;
    for (int i = 0; i < N_PER; ++i) { offs[i] = acc; acc += cur[i]; }
    offs[N_PER] = acc;
  }
  __syncthreads();
  for (int i = tid; i < N_PER; i += NT) {
    const int deg = offs[i + 1] - offs[i];
    invs[i] = rsqrtf((float)deg + 1.f);             // +1 self loop
    cur[i] = offs[i];                               // fill cursor
  }
  __syncthreads();
  // ---- CSR fill: bucket sources by destination ----
  for (int e = tid; e < EPG; e += NT) {
    const int s = esrc[ebase + e] - nbase;
    const int d = edst[ebase + e] - nbase;
    adj[atomicAdd(&cur[d], 1)] = s;
  }
  __syncthreads();

  // ---- GCN layer 1: t = x_g @ W1 (WMMA f32, K=128), gather, tanh ----
  gemm_tiles(x, FIN, (long long)nbase, (long long)N_NODES - 1, FIN, W1, t, tid);
  __syncthreads();
  propagate32(t, h1, invs, offs, adj, b1, tid);

  // ---- layer 2: t = h1 @ W2 (A from LDS, K=32) ----
  gemm_tiles(h1, HS, 0, MPAD - 1, HDIM, W2, t, tid);
  __syncthreads();
  propagate32(t, h2, invs, offs, adj, b2, tid);

  // ---- layer 3 ----
  gemm_tiles(h2, HS, 0, MPAD - 1, HDIM, W3, t, tid);
  __syncthreads();
  propagate32(t, h3, invs, offs, adj, b3, tid);

  // ---- layer 4 (32 -> 1), scalar gather ----
  for (int i = tid; i < N_PER; i += NT) {
    float s = 0.f;
#pragma unroll
    for (int c = 0; c < HDIM; ++c) s += h3[i * HS + c] * W4[c];
    t4[i] = s;
  }
  __syncthreads();
  const float b4v = b4[0];
  for (int i = tid; i < N_PER; i += NT) {
    float acc = 0.f;
    const int beg = offs[i], end = offs[i + 1];
    for (int k = beg; k < end; ++k) {
      const int s = adj[k];
      acc += t4[s] * invs[s];
    }
    const float wi = invs[i];
    h4[i] = tanhf(acc * wi + t4[i] * wi * wi + b4v);
  }
  __syncthreads();

  // ---- SortAggregation: stable descending rank by h4, keep top-30 ----
  for (int i = tid; i < N_PER; i += NT) {
    const float k = h4[i];
    int r = 0;
    for (int j = 0; j < N_PER; ++j) {
      const float kj = h4[j];
      r += (kj > k) || (kj == k && j < i);
    }
    if (r < K_TOP) topidx[r] = i;
  }
  __syncthreads();
  for (int idx = tid; idx < K_TOP * CAT; idx += NT) {
    const int r = idx / CAT, c = idx - r * CAT;
    const int i = topidx[r];
    float v;
    if (c < 32)      v = h1[i * HS + c];
    else if (c < 64) v = h2[i * HS + (c - 32)];
    else if (c < 96) v = h3[i * HS + (c - 64)];
    else             v = h4[i];
    seq[idx] = v;
  }
  __syncthreads();

  // ---- conv1 (k=97, s=97) == per-row 97->16 linear, relu ----
  for (int idx = tid; idx < 16 * 30; idx += NT) {
    const int oc = idx / 30, r = idx - oc * 30;
    float s = cb1[oc];
    const float* w = cw1 + oc * CAT;
    const float* q = seq + r * CAT;
    for (int c = 0; c < CAT; ++c) s += q[c] * w[c];
    c1[oc * 30 + r] = fmaxf(s, 0.f);
  }
  __syncthreads();
  // ---- maxpool 2 ----
  for (int idx = tid; idx < 16 * 15; idx += NT) {
    const int oc = idx / 15, j = idx - oc * 15;
    pl[idx] = fmaxf(c1[oc * 30 + 2 * j], c1[oc * 30 + 2 * j + 1]);
  }
  __syncthreads();
  // ---- conv2 (16ch, k=5) -> [32,11], relu, flattened oc*11+j ----
  for (int idx = tid; idx < 32 * 11; idx += NT) {
    const int oc = idx / 11, j = idx - oc * 11;
    float s = cb2[oc];
    const float* w = cw2 + oc * 80;
    for (int ic = 0; ic < 16; ++ic) {
      const float* pr = pl + ic * 15 + j;
#pragma unroll
      for (int tt = 0; tt < 5; ++tt) s += pr[tt] * w[ic * 5 + tt];
    }
    fl[oc * 11 + j] = fmaxf(s, 0.f);
  }
  __syncthreads();
  // ---- linear 352 -> 128, relu (lw1 stays hot in 192MB L2) ----
  for (int k = tid; k < 128; k += NT) {
    float s = lb1[k];
    for (int f = 0; f < 352; ++f) s += fl[f] * lw1[(long long)f * 128 + k];
    zz[k] = fmaxf(s, 0.f);
  }
  __syncthreads();
  // ---- linear 128 -> 1, sigmoid ----
  if (tid == 0) {
    float s = lb2[0];
    for (int k = 0; k < 128; ++k) s += zz[k] * lw2[k];
    out[g] = 1.f / (1.f + expf(-s));
  }
}

extern "C" void kernel_launch(void* const* d_in, const int* in_sizes, int n_in,
                              void* d_out, int out_size, void* d_ws, size_t ws_size,
                              hipStream_t stream) {
  (void)n_in; (void)d_ws; (void)ws_size; (void)out_size;
  const float* x    = (const float*)d_in[0];
  const int*   ei   = (const int*)  d_in[1];
  // d_in[2] = batch (implicit in the contiguous layout; unused)
  const float* W1   = (const float*)d_in[3];
  const float* b1   = (const float*)d_in[4];
  const float* W2   = (const float*)d_in[5];
  const float* b2   = (const float*)d_in[6];
  const float* W3   = (const float*)d_in[7];
  const float* b3   = (const float*)d_in[8];
  const float* W4   = (const float*)d_in[9];
  const float* b4   = (const float*)d_in[10];
  const float* cw1  = (const float*)d_in[11];
  const float* cb1  = (const float*)d_in[12];
  const float* cw2  = (const float*)d_in[13];
  const float* cb2  = (const float*)d_in[14];
  const float* lw1  = (const float*)d_in[15];
  const float* lb1  = (const float*)d_in[16];
  const float* lw2  = (const float*)d_in[17];
  const float* lb2  = (const float*)d_in[18];
  float* out = (float*)d_out;

  const int nE = in_sizes[1] / 2;                 // 1,600,000
  const int* esrc = ei;
  const int* edst = ei + nE;

  const size_t shmem_dwords =
      (size_t)4 * MPAD * HS       // t, h1, h2, h3 (stride-36 rows)
      + 3 * MPAD                  // h4, t4, invs
      + 2 * MPAD                  // offs, cur
      + EPG                       // adj
      + 32                        // topidx
      + (size_t)K_TOP * CAT       // seq
      + 16 * 30 + 16 * 15 + 352 + 128;
  const size_t shmem = shmem_dwords * sizeof(float);  // ~153 KB -> 2 WGs per WGP

  (void)hipFuncSetAttribute((const void*)dgcnn_fused,
                            hipFuncAttributeMaxDynamicSharedMemorySize,
                            (int)shmem);

  dgcnn_fused<<<dim3(N_GRAPHS), dim3(NT), shmem, stream>>>(
      x, esrc, edst, W1, b1, W2, b2, W3, b3, W4, b4,
      cw1, cb1, cw2, cb2, lw1, lb1, lw2, lb2, out);
}